// TemporalMessagePassingLayer_34600256537308
// MI455X (gfx1250) — compile-verified
//
#include <hip/hip_runtime.h>

// Problem constants (match reference)
#define N_NODES 100000
#define N_EDGES 1000000
#define IN_CH   128
#define OUT_CH  128
#define T_DIM   64

typedef __attribute__((ext_vector_type(2))) float v2f;
typedef __attribute__((ext_vector_type(8))) float v8f;

// -----------------------------------------------------------------------------
// Kernel 1: per-node GEMMs via V_WMMA_F32_16X16X4_F32.
//   blockIdx.y == 0 :  d_out = x @ W_S^T + b_S   (self features, also the
//                      accumulation buffer for the edge scatter-add)
//   blockIdx.y == 1 :  y_T   = x @ W_T^T + b_T   (neighbor transform, ws)
// One wave = 16 node rows x 128 output cols (8 x v8f accumulators).
// A fragment (16x4 f32): lane l -> row l%16, K = k0 + 2*(l/16) + {0,1}
// B fragment (4x16 f32): lane l -> col l%16, K = k0 + 2*(l/16) + {0,1}
//   B(k,n) = W[n][k]  (W is [out,in] row-major -> contiguous float2 in k)
// -----------------------------------------------------------------------------
__global__ __launch_bounds__(256) void node_gemm_kernel(
    const float* __restrict__ x,
    const float* __restrict__ Ws, const float* __restrict__ bS,
    const float* __restrict__ Wt, const float* __restrict__ bT,
    float* __restrict__ outS, float* __restrict__ outT)
{
    const int wave = (blockIdx.x * blockDim.x + threadIdx.x) >> 5;
    const int lane = threadIdx.x & 31;
    const int half = lane >> 4;     // 0: lanes 0-15, 1: lanes 16-31
    const int l16  = lane & 15;
    if (wave >= N_NODES / 16) return;   // uniform per wave: EXEC stays all-1s

    const float* W = blockIdx.y ? Wt   : Ws;
    const float* B = blockIdx.y ? bT   : bS;
    float*       O = blockIdx.y ? outT : outS;

    const int m_base = wave * 16;

    v8f acc[8];
#pragma unroll
    for (int j = 0; j < 8; ++j) {
        const float bv = B[j * 16 + l16];
        acc[j] = (v8f){bv, bv, bv, bv, bv, bv, bv, bv};
    }

    const float* arow = x + (size_t)(m_base + l16) * IN_CH + 2 * half;
#pragma unroll 4
    for (int k0 = 0; k0 < IN_CH; k0 += 4) {
        const v2f a = *(const v2f*)(arow + k0);
#pragma unroll
        for (int j = 0; j < 8; ++j) {
            const v2f b = *(const v2f*)(W + (size_t)(j * 16 + l16) * IN_CH + k0 + 2 * half);
            acc[j] = __builtin_amdgcn_wmma_f32_16x16x4_f32(
                false, a, false, b, (short)0, acc[j], false, false);
        }
    }

    // C/D layout: VGPR r, lanes 0-15 -> M=r, lanes 16-31 -> M=r+8; N = l16.
#pragma unroll
    for (int r = 0; r < 8; ++r) {
        const int m = m_base + r + 8 * half;
#pragma unroll
        for (int j = 0; j < 8; ++j)
            O[(size_t)m * OUT_CH + j * 16 + l16] = acc[j][r];
    }
}

// -----------------------------------------------------------------------------
// Kernel 2: temporal GEMM (1M x 64 @ 64 x 128) via WMMA, fused with the
// per-edge gather of y_T[col] and scatter-add (global_atomic_add_f32) into
// d_out[row]. One wave = 16 edges x 128 output cols.
// -----------------------------------------------------------------------------
__global__ __launch_bounds__(256) void edge_gemm_scatter_kernel(
    const float* __restrict__ tf,      // [N_EDGES, T_DIM]
    const int*   __restrict__ ei,      // [2, N_EDGES]: ei[e]=row(dst), ei[E+e]=col(src)
    const float* __restrict__ Tw,      // [OUT_CH, T_DIM]
    const float* __restrict__ Tb,      // [OUT_CH]
    const float* __restrict__ yT,      // [N_NODES, OUT_CH]
    float*       __restrict__ out)     // [N_NODES, OUT_CH], pre-seeded with self
{
    const int wave = (blockIdx.x * blockDim.x + threadIdx.x) >> 5;
    const int lane = threadIdx.x & 31;
    const int half = lane >> 4;
    const int l16  = lane & 15;
    if (wave >= N_EDGES / 16) return;   // uniform per wave

    const int e_base = wave * 16;

    v8f acc[8];
#pragma unroll
    for (int j = 0; j < 8; ++j) {
        const float bv = Tb[j * 16 + l16];
        acc[j] = (v8f){bv, bv, bv, bv, bv, bv, bv, bv};
    }

    const float* arow = tf + (size_t)(e_base + l16) * T_DIM + 2 * half;
#pragma unroll 4
    for (int k0 = 0; k0 < T_DIM; k0 += 4) {
        const v2f a = *(const v2f*)(arow + k0);
#pragma unroll
        for (int j = 0; j < 8; ++j) {
            const v2f b = *(const v2f*)(Tw + (size_t)(j * 16 + l16) * T_DIM + k0 + 2 * half);
            acc[j] = __builtin_amdgcn_wmma_f32_16x16x4_f32(
                false, a, false, b, (short)0, acc[j], false, false);
        }
    }

    // message = temporal_encoding + y_T[col];  out[row] += message (atomic)
#pragma unroll
    for (int r = 0; r < 8; ++r) {
        const int e   = e_base + r + 8 * half;
        const int dst = ei[e];
        const int src = ei[N_EDGES + e];
        const float* yrow = yT  + (size_t)src * OUT_CH;
        float*       orow = out + (size_t)dst * OUT_CH;
#pragma unroll
        for (int j = 0; j < 8; ++j) {
            const int n = j * 16 + l16;
            atomicAdd(&orow[n], acc[j][r] + yrow[n]);
        }
    }
}

// -----------------------------------------------------------------------------
// Kernel 3: in-place ReLU over the accumulated output.
// -----------------------------------------------------------------------------
__global__ void relu_kernel(float* __restrict__ out, int n)
{
    const int i = blockIdx.x * blockDim.x + threadIdx.x;
    if (i < n) {
        const float v = out[i];
        out[i] = v > 0.0f ? v : 0.0f;
    }
}

extern "C" void kernel_launch(void* const* d_in, const int* in_sizes, int n_in,
                              void* d_out, int out_size, void* d_ws, size_t ws_size,
                              hipStream_t stream)
{
    const float* x    = (const float*)d_in[0];
    const int*   ei   = (const int*)  d_in[1];   // edge_index [2, N_EDGES]
    const float* tf   = (const float*)d_in[2];   // temporal_features
    const float* WSw  = (const float*)d_in[3];
    const float* WSb  = (const float*)d_in[4];
    const float* WTw  = (const float*)d_in[5];
    const float* WTb  = (const float*)d_in[6];
    const float* Tw   = (const float*)d_in[7];
    const float* Tb   = (const float*)d_in[8];

    float* out = (float*)d_out;                  // [N_NODES, OUT_CH]
    float* yT  = (float*)d_ws;                   // [N_NODES, OUT_CH] scratch (51.2 MB)

    // 1) node transforms: d_out = self features, yT = neighbor transform
    {
        dim3 block(256);                               // 8 waves
        dim3 grid((N_NODES / 16 + 7) / 8, 2);
        node_gemm_kernel<<<grid, block, 0, stream>>>(x, WSw, WSb, WTw, WTb, out, yT);
    }
    // 2) temporal GEMM + gather + scatter-add into d_out
    {
        dim3 block(256);
        dim3 grid((N_EDGES / 16 + 7) / 8);
        edge_gemm_scatter_kernel<<<grid, block, 0, stream>>>(tf, ei, Tw, Tb, yT, out);
    }
    // 3) ReLU in place
    {
        const int n = N_NODES * OUT_CH;
        relu_kernel<<<(n + 255) / 256, 256, 0, stream>>>(out, n);
    }
}